// Speller_63007170232648
// MI455X (gfx1250) — compile-verified
//
#include <hip/hip_runtime.h>
#include <cstddef>
#include <cstdint>

#define B_     64
#define T_     2000
#define HC_    512
#define HS_    512
#define V_     128
#define HIN_   640
#define G4_    2048          // 4*HS
#define KTOT_  1152          // HIN + HS
#define NSTEP_ 100
#define TILE_T_ 20           // rows of h per TDM tile
#define NT_    100           // tiles per batch row (2000/20)
#define NTPB_  5             // tiles per block (double-buffered pipeline)
#define NBLK_  20            // blocks per batch row
#define EPS_   1e-5f

static_assert(TILE_T_ * NT_ == T_, "tiling");
static_assert(NTPB_ * NBLK_ == NT_, "tiling");

typedef __attribute__((ext_vector_type(2))) float v2f;
typedef __attribute__((ext_vector_type(8))) float v8f;
typedef __attribute__((ext_vector_type(4))) unsigned v4u;
typedef __attribute__((ext_vector_type(8))) int v8i;
typedef __attribute__((ext_vector_type(4))) int v4i;

#if __has_builtin(__builtin_amdgcn_tensor_load_to_lds)
#define HAVE_TDM_ 1
#else
#define HAVE_TDM_ 0
#endif

// ---------------- workspace layout (float offsets) ----------------
static const size_t OFF_X     = 0;                        // 64*640
static const size_t OFF_A     = OFF_X     + 40960;        // 64*1152
static const size_t OFF_HSA   = OFF_A     + 73728;        // 64*512
static const size_t OFF_HSB   = OFF_HSA   + 32768;        // 64*512
static const size_t OFF_CS    = OFF_HSB   + 32768;        // 64*512
static const size_t OFF_GATES = OFF_CS    + 32768;        // 64*2048
static const size_t OFF_CTX   = OFF_GATES + 131072;       // 64*512
static const size_t OFF_BNA   = OFF_CTX   + 32768;        // 640
static const size_t OFF_BNB   = OFF_BNA   + 640;          // 640
static const size_t OFF_BCAT  = OFF_BNB   + 640;          // 2048
static const size_t OFF_MBLK  = OFF_BCAT  + 2048;         // 64*20
static const size_t OFF_SBLK  = OFF_MBLK  + 1280;         // 64*20
static const size_t OFF_SC    = OFF_SBLK  + 1280;         // 64*100
static const size_t OFF_MW    = OFF_SC    + 6400;         // 64*100
static const size_t OFF_PREDS = OFF_MW    + 6400;         // 100*64 ints
static const size_t OFF_WCAT  = OFF_PREDS + 6400;         // 2048*1152
static const size_t OFF_CTXP  = OFF_WCAT  + 2359296;      // 64*20*512

// d_out layout (floats): yhat[64*100*128] | seq[64 int32] | atts[100*64*2000]
static const size_t OUT_SEQ   = 819200;
static const size_t OUT_ATT   = 819264;

// ---------------- TDM helper: 2-D f32 tile (TILE_T_ x 512) -> LDS ----------------
__device__ __forceinline__ void tdm_load_tile(const float* gptr, unsigned lds_off) {
#if HAVE_TDM_
  unsigned long long ga = (unsigned long long)(uintptr_t)gptr;
  v4u g0;
  g0[0] = 1u;                                          // count=1, user descriptor
  g0[1] = lds_off;                                     // lds_addr (bytes)
  g0[2] = (unsigned)(ga & 0xFFFFFFFFu);                // global_addr[31:0]
  g0[3] = (unsigned)((ga >> 32) & 0x1FFFFFFu)          // global_addr[56:32]
        | (2u << 30);                                  // type = 2 ("image")
  v8i g1;
  g1[0] = (int)(2u << 16);                             // data_size = 4B
  g1[1] = (int)((unsigned)HC_ << 16);                  // tensor_dim0[15:0] (=512)
  g1[2] = (int)(((unsigned)HC_ >> 16) |                // tensor_dim0[31:16]
                ((unsigned)(T_ & 0xFFFF) << 16));      // tensor_dim1[15:0]
  g1[3] = (int)(((unsigned)T_ >> 16) |                 // tensor_dim1[31:16]
                ((unsigned)HC_ << 16));                // tile_dim0 = 512
  g1[4] = (int)TILE_T_;                                // tile_dim1 = 20, tile_dim2 = 0
  g1[5] = (int)HC_;                                    // tensor_dim0_stride[31:0] = 512
  g1[6] = 0;                                           // stride hi / dim1_stride lo
  g1[7] = 0;                                           // dim1_stride hi
  v4i gz4 = {0, 0, 0, 0};
  v8i gz8 = {0, 0, 0, 0, 0, 0, 0, 0};
  // amdgpu-toolchain (clang-23) 6-arg form:
  // (uint32x4 g0, int32x8 g1, int32x4, int32x4, int32x8, i32 cpol)
  __builtin_amdgcn_tensor_load_to_lds(g0, g1, gz4, gz4, gz8, 0);
#else
  (void)gptr; (void)lds_off;
#endif
}

// ---------------- init kernels ----------------
__global__ void k_build_w(const float* __restrict__ Wih, const float* __restrict__ Whh,
                          const float* __restrict__ bih, const float* __restrict__ bhh,
                          float* __restrict__ Wcat, float* __restrict__ bcat) {
  size_t idx = (size_t)blockIdx.x * 256 + threadIdx.x;  // over 2048*1152
  if (idx < (size_t)G4_ * KTOT_) {
    int n = (int)(idx / KTOT_), k = (int)(idx % KTOT_);
    Wcat[idx] = (k < HIN_) ? Wih[(size_t)n * HIN_ + k] : Whh[(size_t)n * HS_ + (k - HIN_)];
  }
  if (idx < G4_) bcat[idx] = bih[idx] + bhh[idx];
}

__global__ void k_init_x(const float* __restrict__ h, float* __restrict__ x) {
  int b = blockIdx.x, j = threadIdx.x;                   // <<<64, 640>>>
  float v;
  if (j < V_) v = (j == V_ - 2) ? 1.0f : 0.0f;           // one-hot SOS (V-2)
  else        v = h[(size_t)b * T_ * HC_ + (j - V_)];    // h[b, 0, :]
  x[b * HIN_ + j] = v;
}

__global__ void k_zero(float* __restrict__ p, int n) {
  int i = blockIdx.x * 256 + threadIdx.x;
  if (i < n) p[i] = 0.0f;
}

// ---------------- per-step kernels ----------------
__global__ void k_bnstats(const float* __restrict__ x, const float* __restrict__ gamma,
                          const float* __restrict__ beta, float* __restrict__ bnA,
                          float* __restrict__ bnB) {
  int j = blockIdx.x * 128 + threadIdx.x;                // <<<5, 128>>> over 640 cols
  if (j >= HIN_) return;
  float s = 0.f, s2 = 0.f;
  for (int b = 0; b < B_; ++b) { float v = x[b * HIN_ + j]; s += v; s2 += v * v; }
  float mu = s * (1.0f / B_);
  float var = s2 * (1.0f / B_) - mu * mu;
  float a = gamma[j] * rsqrtf(var + EPS_);
  bnA[j] = a;
  bnB[j] = beta[j] - mu * a;
}

__global__ void k_build_a(const float* __restrict__ x, const float* __restrict__ bnA,
                          const float* __restrict__ bnB, const float* __restrict__ hs,
                          float* __restrict__ A) {
  int idx = blockIdx.x * 256 + threadIdx.x;              // <<<288, 256>>> over 73728
  if (idx >= B_ * KTOT_) return;
  int b = idx / KTOT_, k = idx % KTOT_;
  A[idx] = (k < HIN_) ? (x[b * HIN_ + k] * bnA[k] + bnB[k]) : hs[b * HS_ + (k - HIN_)];
}

// gates = A(64x1152) @ Wcat^T + bcat via V_WMMA_F32_16X16X4_F32; one wave / 16x16 tile.
__global__ void k_gates(const float* __restrict__ A, const float* __restrict__ W,
                        const float* __restrict__ bias, float* __restrict__ gates) {
  int wid  = blockIdx.x * 4 + (threadIdx.x >> 5);        // <<<128, 128>>>
  int lane = threadIdx.x & 31;
  int mt = wid & 3, nt = wid >> 2;
  int l15 = lane & 15;
  int kh2 = (lane >> 4) * 2;
  const float* pa = A + (size_t)(mt * 16 + l15) * KTOT_ + kh2;
  const float* pb = W + (size_t)(nt * 16 + l15) * KTOT_ + kh2;
  v8f acc = {0.f, 0.f, 0.f, 0.f, 0.f, 0.f, 0.f, 0.f};
  for (int kb = 0; kb < KTOT_; kb += 4) {
    v2f av = *(const v2f*)(pa + kb);
    v2f bv = *(const v2f*)(pb + kb);
    acc = __builtin_amdgcn_wmma_f32_16x16x4_f32(false, av, false, bv,
                                                (short)0, acc, false, false);
  }
  int n = nt * 16 + l15;
  float bs = bias[n];
  int mbase = mt * 16 + (lane >> 4) * 8;
#pragma unroll
  for (int r = 0; r < 8; ++r)
    gates[(size_t)(mbase + r) * G4_ + n] = acc[r] + bs;
}

__global__ void k_lstm(const float* __restrict__ gates, float* __restrict__ cs,
                       float* __restrict__ hout) {
  int idx = blockIdx.x * 256 + threadIdx.x;              // <<<128, 256>>> over 32768
  int b = idx >> 9, k = idx & 511;
  const float* g = gates + (size_t)b * G4_;
  float gi = g[k], gf = g[k + 512], gg = g[k + 1024], go = g[k + 1536];
  float si = 1.f / (1.f + __expf(-gi));
  float sf = 1.f / (1.f + __expf(-gf));
  float so = 1.f / (1.f + __expf(-go));
  float c = sf * cs[idx] + si * tanhf(gg);
  cs[idx] = c;
  hout[idx] = so * tanhf(c);
}

// Flash-attention pass with TDM double buffering: each block walks NTPB_ tiles,
// DMA-ing tile i+1 into the idle LDS buffer while computing on tile i.
// Per block outputs: running (m, s), partial ctx; per tile: write-time max.
__global__ void k_attn(const float* __restrict__ h, const float* __restrict__ hn,
                       float* __restrict__ att, float* __restrict__ mblk,
                       float* __restrict__ sblk, float* __restrict__ ctxp,
                       float* __restrict__ mwrite) {
  __shared__ float tileH[2][TILE_T_ * HC_];              // 2 x 40 KB
  __shared__ float shHn[HS_];
  __shared__ float shE[TILE_T_];
  __shared__ float shP[TILE_T_];
  __shared__ float shMold, shMnew;
  int tid = threadIdx.x;                                 // <<<dim3(20,64), 256>>>
  int b = blockIdx.y, blk = blockIdx.x;
  int g0 = blk * NTPB_;
  const float* hb = h + ((size_t)b * T_ + (size_t)g0 * TILE_T_) * HC_;
  unsigned lds0 = (unsigned)(uintptr_t)(void*)&tileH[0][0];
  unsigned lds1 = (unsigned)(uintptr_t)(void*)&tileH[1][0];

  shHn[tid]       = hn[b * HS_ + tid];
  shHn[tid + 256] = hn[b * HS_ + tid + 256];

#if HAVE_TDM_
  if (tid < 32) tdm_load_tile(hb, lds0);                 // wave-level TDM issue
#endif

  int w = tid >> 5, lane = tid & 31;
  float mrun = -__builtin_inff(), srun = 0.f;            // owned by thread 0
  float c0 = 0.f, c1 = 0.f;                              // per-thread ctx accum

#pragma unroll
  for (int it = 0; it < NTPB_; ++it) {
    int g = g0 + it;
    float* tile = &tileH[it & 1][0];
#if HAVE_TDM_
    if (tid < 32) {
      if (it + 1 < NTPB_) {
        tdm_load_tile(hb + (size_t)(it + 1) * TILE_T_ * HC_,
                      ((it + 1) & 1) ? lds1 : lds0);
        __builtin_amdgcn_s_wait_tensorcnt(1);            // tile 'it' done, 'it+1' in flight
      } else {
        __builtin_amdgcn_s_wait_tensorcnt(0);            // last tile done
      }
    }
    __syncthreads();                                     // tile 'it' visible to block
#else
    {
      const float4* s4 = (const float4*)(hb + (size_t)it * TILE_T_ * HC_);
      float4* d4 = (float4*)tile;
      for (int i = tid; i < TILE_T_ * HC_ / 4; i += 256) d4[i] = s4[i];
    }
    __syncthreads();
#endif
    // e[t] = dot(h_new, h[t]) — one wave per row
    for (int t = w; t < TILE_T_; t += 8) {
      float acc = 0.f;
#pragma unroll
      for (int j = 0; j < HC_ / 32; ++j) {
        int k = lane + j * 32;
        acc += shHn[k] * tile[t * HC_ + k];
      }
      for (int off = 16; off; off >>= 1) acc += __shfl_xor(acc, off, 32);
      if (lane == 0) shE[t] = acc;
    }
    __syncthreads();
    if (tid == 0) {
      float mt = shE[0];
      for (int t = 1; t < TILE_T_; ++t) mt = fmaxf(mt, shE[t]);
      float mnew = fmaxf(mrun, mt);
      shMold = mrun; shMnew = mnew;
      mrun = mnew;
      mwrite[b * NT_ + g] = mnew;                        // base used for atts numerators
    }
    __syncthreads();
    float fac = __expf(shMold - shMnew);                 // 0 on first tile (-inf base)
    c0 *= fac; c1 *= fac;
    if (tid < TILE_T_) {
      float p = __expf(shE[tid] - shMnew);
      shP[tid] = p;
      att[(size_t)b * T_ + (size_t)g * TILE_T_ + tid] = p;  // rescaled later
    }
    __syncthreads();
    if (tid == 0) {
      float s = 0.f;
      for (int t = 0; t < TILE_T_; ++t) s += shP[t];
      srun = srun * fac + s;
    }
    for (int t = 0; t < TILE_T_; ++t) {
      float p = shP[t];
      c0 += p * tile[t * HC_ + tid];
      c1 += p * tile[t * HC_ + tid + 256];
    }
    __syncthreads();                                     // done reading this buffer
  }
  if (tid == 0) { mblk[b * NBLK_ + blk] = mrun; sblk[b * NBLK_ + blk] = srun; }
  ctxp[((size_t)b * NBLK_ + blk) * HC_ + tid]       = c0;
  ctxp[((size_t)b * NBLK_ + blk) * HC_ + tid + 256] = c1;
}

__global__ void k_attn_reduce(const float* __restrict__ mblk, const float* __restrict__ sblk,
                              const float* __restrict__ ctxp, const float* __restrict__ mwrite,
                              float* __restrict__ ctx, float* __restrict__ scales) {
  __shared__ float shW[NBLK_];
  __shared__ float shM, shS;
  int b = blockIdx.x, tid = threadIdx.x;                 // <<<64, 256>>>
  if (tid == 0) {
    float m = mblk[b * NBLK_];
    for (int j = 1; j < NBLK_; ++j) m = fmaxf(m, mblk[b * NBLK_ + j]);
    float s = 0.f;
    for (int j = 0; j < NBLK_; ++j) {
      float wv = __expf(mblk[b * NBLK_ + j] - m);
      shW[j] = wv;
      s += wv * sblk[b * NBLK_ + j];
    }
    shM = m; shS = s;
  }
  __syncthreads();
  float inv = 1.0f / shS;
  if (tid < NT_) scales[b * NT_ + tid] = __expf(mwrite[b * NT_ + tid] - shM) * inv;
  float c0 = 0.f, c1 = 0.f;
  for (int j = 0; j < NBLK_; ++j) {
    float wv = shW[j];
    c0 += wv * ctxp[((size_t)b * NBLK_ + j) * HC_ + tid];
    c1 += wv * ctxp[((size_t)b * NBLK_ + j) * HC_ + tid + 256];
  }
  ctx[b * HC_ + tid]       = c0 * inv;
  ctx[b * HC_ + tid + 256] = c1 * inv;
}

__global__ void k_attn_scale(float* __restrict__ att, const float* __restrict__ scales) {
  int b = blockIdx.y;                                    // <<<dim3(8,64), 256>>>
  int t = blockIdx.x * 256 + threadIdx.x;
  if (t < T_) att[(size_t)b * T_ + t] *= scales[b * NT_ + t / TILE_T_];
}

// logits = [h_new|ctx] @ Wcd^T + bcd; softmax; write y_hat, x_next, argmax pred.
__global__ void k_decode(const float* __restrict__ hn, const float* __restrict__ ctx,
                         const float* __restrict__ Wcd, const float* __restrict__ bcd,
                         float* __restrict__ yhat, float* __restrict__ x,
                         int* __restrict__ preds, int s) {
  __shared__ float shH[HS_], shC[HS_];
  __shared__ float red[V_];
  __shared__ int redi[V_];
  int b = blockIdx.x, v = threadIdx.x;                   // <<<64, 128>>>
#pragma unroll
  for (int i = 0; i < 4; ++i) {
    shH[v + 128 * i] = hn[b * HS_ + v + 128 * i];
    shC[v + 128 * i] = ctx[b * HC_ + v + 128 * i];
  }
  __syncthreads();
  const float* wr = Wcd + (size_t)v * (HS_ + HC_);
  float acc = bcd[v];
  for (int j = 0; j < HS_; ++j) acc += shH[j] * wr[j];
  for (int j = 0; j < HC_; ++j) acc += shC[j] * wr[HS_ + j];
  red[v] = acc; redi[v] = v;
  __syncthreads();
  for (int off = 64; off >= 1; off >>= 1) {
    if (v < off) {
      float o = red[v + off]; int oi = redi[v + off];
      if (o > red[v] || (o == red[v] && oi < redi[v])) { red[v] = o; redi[v] = oi; }
    }
    __syncthreads();
  }
  float M = red[0]; int amax = redi[0];
  __syncthreads();
  float p = __expf(acc - M);
  red[v] = p;
  __syncthreads();
  for (int off = 64; off >= 1; off >>= 1) {
    if (v < off) red[v] += red[v + off];
    __syncthreads();
  }
  float y = p / red[0];
  yhat[(size_t)b * NSTEP_ * V_ + (size_t)s * V_ + v] = y;
  x[b * HIN_ + v] = y;
#pragma unroll
  for (int i = 0; i < 4; ++i)
    x[b * HIN_ + V_ + v * 4 + i] = shC[v * 4 + i];
  if (v == 0) preds[s * B_ + b] = amax;
}

__global__ void k_seqlen(const int* __restrict__ preds, int* __restrict__ seq) {
  int b = threadIdx.x;                                   // <<<1, 64>>>
  int r = NSTEP_;
  for (int s = 0; s < NSTEP_; ++s)
    if (preds[s * B_ + b] == V_ - 1) { r = s; break; }
  seq[b] = r;
}

// ---------------- launcher ----------------
extern "C" void kernel_launch(void* const* d_in, const int* in_sizes, int n_in,
                              void* d_out, int out_size, void* d_ws, size_t ws_size,
                              hipStream_t stream) {
  const float* h     = (const float*)d_in[0];
  const float* gamma = (const float*)d_in[1];
  const float* beta  = (const float*)d_in[2];
  const float* Wih   = (const float*)d_in[3];
  const float* Whh   = (const float*)d_in[4];
  const float* bih   = (const float*)d_in[5];
  const float* bhh   = (const float*)d_in[6];
  const float* Wcd   = (const float*)d_in[7];
  const float* bcd   = (const float*)d_in[8];
  float* out = (float*)d_out;
  float* ws  = (float*)d_ws;

  float* x      = ws + OFF_X;
  float* Abuf   = ws + OFF_A;
  float* hsA    = ws + OFF_HSA;
  float* hsB    = ws + OFF_HSB;
  float* cs     = ws + OFF_CS;
  float* gates  = ws + OFF_GATES;
  float* ctx    = ws + OFF_CTX;
  float* bnA    = ws + OFF_BNA;
  float* bnB    = ws + OFF_BNB;
  float* bcat   = ws + OFF_BCAT;
  float* mblk   = ws + OFF_MBLK;
  float* sblk   = ws + OFF_SBLK;
  float* scales = ws + OFF_SC;
  float* mwrite = ws + OFF_MW;
  int*   preds  = (int*)(ws + OFF_PREDS);
  float* Wcat   = ws + OFF_WCAT;
  float* ctxp   = ws + OFF_CTXP;

  float* yhat = out;
  int*   seq  = (int*)(out + OUT_SEQ);
  float* atts = out + OUT_ATT;

  k_build_w<<<9216, 256, 0, stream>>>(Wih, Whh, bih, bhh, Wcat, bcat);
  k_init_x<<<B_, HIN_, 0, stream>>>(h, x);
  k_zero<<<384, 256, 0, stream>>>(hsA, 3 * B_ * HS_);    // hsA, hsB, cs (contiguous)

  for (int s = 0; s < NSTEP_; ++s) {
    float* hin   = (s & 1) ? hsB : hsA;
    float* hout  = (s & 1) ? hsA : hsB;
    float* att_s = atts + (size_t)s * B_ * T_;
    k_bnstats<<<5, 128, 0, stream>>>(x, gamma, beta, bnA, bnB);
    k_build_a<<<288, 256, 0, stream>>>(x, bnA, bnB, hin, Abuf);
    k_gates<<<128, 128, 0, stream>>>(Abuf, Wcat, bcat, gates);
    k_lstm<<<128, 256, 0, stream>>>(gates, cs, hout);
    k_attn<<<dim3(NBLK_, B_), 256, 0, stream>>>(h, hout, att_s, mblk, sblk, ctxp, mwrite);
    k_attn_reduce<<<B_, 256, 0, stream>>>(mblk, sblk, ctxp, mwrite, ctx, scales);
    k_attn_scale<<<dim3(8, B_), 256, 0, stream>>>(att_s, scales);
    k_decode<<<B_, V_, 0, stream>>>(hout, ctx, Wcd, bcd, yhat, x, preds, s);
  }
  k_seqlen<<<1, B_, 0, stream>>>(preds, seq);
}